// BeamSplitterUnitaryLayer_16020228014172
// MI455X (gfx1250) — compile-verified
//
#include <hip/hip_runtime.h>
#include <hip/hip_bf16.h>
#include <math.h>

// ---------------------------------------------------------------------------
// BeamSplitterUnitaryLayer for MI455X (gfx1250).
//
// y[row, 2g:2g+2] = M[g] @ x[row, 2g:2g+2],  M[g] = R(phase[g]) * polar(W[g])
//
// Roofline: 512 MB traffic, 0.27 GFLOP -> 0.5 FLOP/byte -> pure HBM-bound
// (~22 us at 23.3 TB/s). Strategy:
//   * polar(2x2) has a closed form (no SVD): det>0 -> nearest rotation
//     [[a+d, -(c-b)],[c-b, a+d]]/h ; det<0 -> nearest reflection.
//   * stage W/phase and the streaming x tiles straight into LDS with the
//     gfx1250 async global->LDS DMA (ASYNCcnt path, inline asm = portable
//     across ROCm7.2 / amdgpu-toolchain).
//   * apply the block-diagonal 2x2s with v_wmma_f32_16x16x4_f32 (f32 kept
//     end-to-end): A = 16(row)x4(chan) x-tile from LDS, B = 4x16 slice of the
//     block-diagonal M built per-lane with cndmasks, 4 WMMAs per 16x16 tile.
// ---------------------------------------------------------------------------

typedef float v2f __attribute__((ext_vector_type(2)));
typedef float v8f __attribute__((ext_vector_type(8)));

#define BSU_B 8
#define BSU_T 4096
#define BSU_C 2048
#define BSU_G (BSU_C / 2)          // 1024 pair-groups
#define BSU_ROWS (BSU_B * BSU_T)   // 32768 rows
#define TPB 256
#define WAVES_PER_BLOCK 8
#define ROWS_PER_WAVE 16
#define ROWS_PER_BLOCK (WAVES_PER_BLOCK * ROWS_PER_WAVE)  // 128
#define NBLOCKS (BSU_ROWS / ROWS_PER_BLOCK)               // 256
#define NCT (BSU_C / 16)                                  // 128 channel tiles
// LDS x-tile row stride in floats: even (b64-aligned A reads), multiple of 4
// (b128-aligned row starts), and 20*m mod 64 hits 16 distinct banks.
#define XSTRIDE 20

// Low 32 bits of a generic pointer to LDS == wave-relative LDS byte address.
__device__ __forceinline__ unsigned lds_off(const void* p) {
  return (unsigned)(uintptr_t)p;
}

// gfx1250 async DMA: 16 bytes global -> LDS per lane. Tracked by ASYNCcnt.
__device__ __forceinline__ void async_copy_b128(unsigned lds_byte, const void* gaddr) {
  asm volatile("global_load_async_to_lds_b128 %0, %1, off"
               :: "v"(lds_byte), "v"(gaddr) : "memory");
}
__device__ __forceinline__ void wait_async0() {
  asm volatile("s_wait_asynccnt 0" ::: "memory");
}

__global__ __launch_bounds__(TPB) void bsu_apply_kernel(
    const float* __restrict__ x,
    const float* __restrict__ Wp,
    const float* __restrict__ phase,
    float* __restrict__ y)
{
  // W_pairs staged here, then overwritten in place by composed M (16 KB).
  __shared__ __align__(16) float ldsM[BSU_G * 4];
  __shared__ __align__(16) float ldsPh[BSU_G];
  __shared__ __align__(16) float ldsX[WAVES_PER_BLOCK][ROWS_PER_WAVE * XSTRIDE];

  const int tid  = threadIdx.x;
  const int lane = tid & 31;
  const int wv   = tid >> 5;

  // ---- Stage W_pairs (16 KB) + phase (4 KB) via async global->LDS DMA ----
  for (int i = tid; i < BSU_G; i += TPB)
    async_copy_b128(lds_off(&ldsM[i * 4]), Wp + (size_t)i * 4);
  for (int i = tid; i < BSU_G / 4; i += TPB)
    async_copy_b128(lds_off(&ldsPh[i * 4]), phase + (size_t)i * 4);
  wait_async0();
  __syncthreads();

  // ---- Compose M[g] = R(phase) * polar(W) in place (closed-form polar) ----
  #pragma unroll
  for (int k = 0; k < BSU_G / TPB; ++k) {
    const int g = tid + k * TPB;
    float4 w = *(float4*)&ldsM[g * 4];  // (w00, w01, w10, w11)
    const float det = w.x * w.w - w.y * w.z;
    float q00, q01, q10, q11;
    if (det >= 0.f) {                   // nearest rotation
      const float a = w.x + w.w, gmm = w.z - w.y;
      const float h = rsqrtf(a * a + gmm * gmm);
      q00 = a * h;  q01 = -gmm * h;  q10 = gmm * h;  q11 = a * h;
    } else {                            // nearest reflection
      const float mu = w.x - w.w, nu = w.y + w.z;
      const float h = rsqrtf(mu * mu + nu * nu);
      q00 = mu * h;  q01 = nu * h;  q10 = nu * h;  q11 = -mu * h;
    }
    float sp, cp;
    __sincosf(ldsPh[g], &sp, &cp);
    float4 m;
    m.x = cp * q00 - sp * q10;  m.y = cp * q01 - sp * q11;
    m.z = sp * q00 + cp * q10;  m.w = sp * q01 + cp * q11;
    *(float4*)&ldsM[g * 4] = m;
  }
  __syncthreads();

  // ---- Per-lane WMMA layout constants (ISA 16x4 A / 4x16 B / 16x16 D) ----
  const int  n     = lane & 15;        // A row m / B+D column n
  const bool hi    = lane >= 16;
  const int  k0    = hi ? 2 : 0;       // this lane's K pair within a chunk
  const int  kcsel = n >> 2;           // K-chunk holding column n's 2x2 block
  const bool par   = (((n >> 1) & 1) == (hi ? 1 : 0));

  const long rowBase = (long)(blockIdx.x * WAVES_PER_BLOCK + wv) * ROWS_PER_WAVE;
  const unsigned xbase = lds_off(&ldsX[wv][0]);

  // Coalesced tile fetch pattern: 64 float4s (16 rows x 16 ch), 2 per lane.
  const int idx0 = lane, idx1 = lane + 32;
  const int r0 = idx0 >> 2, c40 = idx0 & 3;
  const int r1 = idx1 >> 2, c41 = idx1 & 3;
  const unsigned l0 = xbase + (unsigned)(r0 * XSTRIDE + c40 * 4) * 4u;
  const unsigned l1 = xbase + (unsigned)(r1 * XSTRIDE + c41 * 4) * 4u;
  const float* g0 = x + (rowBase + r0) * BSU_C + c40 * 4;
  const float* g1 = x + (rowBase + r1) * BSU_C + c41 * 4;
  float* ybase = y + (rowBase + (hi ? 8 : 0)) * BSU_C + n;

  for (int ct = 0; ct < NCT; ++ct) {
    const int chBase = ct * 16;

    // Async-DMA this wave's 16x16 x-tile into LDS (coalesced 128b loads).
    async_copy_b128(l0, g0 + chBase);
    async_copy_b128(l1, g1 + chBase);

    // This lane's M row: output channel n belongs to group g, row (n&1).
    const int grp = (chBase + n) >> 1;
    const v2f mrow = *(const v2f*)&ldsM[grp * 4 + (n & 1) * 2];

    wait_async0();

    v8f acc = {};
    #pragma unroll
    for (int kc = 0; kc < 4; ++kc) {
      // A: x[row = n][chan = chBase + 4*kc + k0 .. +1] from LDS (bank-free).
      const v2f a = *(const v2f*)&ldsX[wv][n * XSTRIDE + kc * 4 + k0];
      // B: 4x16 block-diagonal slice of M; nonzero only when this lane's
      // K pair coincides with column n's 2x2 block.
      const bool on = par && (kcsel == kc);
      v2f b;
      b.x = on ? mrow.x : 0.f;
      b.y = on ? mrow.y : 0.f;
      acc = __builtin_amdgcn_wmma_f32_16x16x4_f32(
          /*neg_a=*/false, a, /*neg_b=*/false, b,
          /*c_mod=*/(short)0, acc, /*reuse_a=*/false, /*reuse_b=*/false);
    }

    // D layout: VGPR r, lanes 0-15 -> row r, lanes 16-31 -> row r+8; N = n.
    float* yo = ybase + chBase;
    #pragma unroll
    for (int r = 0; r < 8; ++r)
      yo[(long)r * BSU_C] = acc[r];
  }
}

extern "C" void kernel_launch(void* const* d_in, const int* in_sizes, int n_in,
                              void* d_out, int out_size, void* d_ws, size_t ws_size,
                              hipStream_t stream) {
  (void)in_sizes; (void)n_in; (void)d_ws; (void)ws_size; (void)out_size;
  const float* x  = (const float*)d_in[0];
  const float* Wp = (const float*)d_in[1];
  const float* ph = (const float*)d_in[2];
  float* y = (float*)d_out;
  bsu_apply_kernel<<<NBLOCKS, TPB, 0, stream>>>(x, Wp, ph, y);
}